// Decoder_72911364816952
// MI455X (gfx1250) — compile-verified
//
#include <hip/hip_runtime.h>
#include <math.h>

typedef __attribute__((ext_vector_type(2))) float v2f;
typedef __attribute__((ext_vector_type(8))) float v8f;

#define WPB 8  // waves per block (wave32 -> 256 threads)

__device__ __forceinline__ v8f wmma4(v2f a, v2f b, v8f c) {
  // V_WMMA_F32_16X16X4_F32: D(16x16) = A(16x4) * B(4x16) + C
  return __builtin_amdgcn_wmma_f32_16x16x4_f32(false, a, false, b, (short)0, c,
                                               false, false);
}

// ---------------------------------------------------------------------------
// lat_new = lat @ frac_W[l] + frac_b[l]   (4x256 @ 256x256)
__global__ void __launch_bounds__(256) latnewk(const float* __restrict__ lat,
                                               const float* __restrict__ fw,
                                               const float* __restrict__ fb,
                                               float* __restrict__ ln) {
  int j = threadIdx.x;  // 256 columns
  float a0 = 0.f, a1 = 0.f, a2 = 0.f, a3 = 0.f;
  for (int k = 0; k < 256; ++k) {
    float w = fw[k * 256 + j];
    a0 += lat[k] * w;
    a1 += lat[256 + k] * w;
    a2 += lat[512 + k] * w;
    a3 += lat[768 + k] * w;
  }
  float bb = fb[j];
  ln[j] = a0 + bb; ln[256 + j] = a1 + bb;
  ln[512 + j] = a2 + bb; ln[768 + j] = a3 + bb;
}

// ---------------------------------------------------------------------------
// Wgen[b][n] = sum_k lat_new[b][k] * Ww[k][n]  (M=4 padded to 16, K=256, N big)
// Bandwidth-bound stream of Ww. 16x32 output tile per wave (2 WMMAs / A-frag),
// software-pipelined loads, pointer-increment addressing, LDS A padded to 16
// rows so the ds_load_b64 needs no EXEC divergence.
__global__ void __launch_bounds__(256) weightgen(const float* __restrict__ Ww,
                                                 const float* __restrict__ latnew,
                                                 float* __restrict__ Wgen, int N) {
  __shared__ float lds[4096];  // 16 x 256, rows 4..15 zero
  for (int i = threadIdx.x; i < 4096; i += 256)
    lds[i] = (i < 1024) ? latnew[i] : 0.f;
  __syncthreads();
  int lane = threadIdx.x & 31, wave = threadIdx.x >> 5;
  int hi = lane >> 4, m = lane & 15;
  long n0 = ((long)blockIdx.x * WPB + wave) << 5;  // 32 columns per wave
  if (n0 >= N) return;  // wave-uniform
  const size_t rs = (size_t)N;                        // row stride (floats)
  const float* r0 = Ww + (size_t)(2 * hi) * rs + n0 + m;  // row 2*hi
  const float* r1 = r0 + rs;                              // row 2*hi+1
  const int abase = m * 256 + 2 * hi;
  v8f acc0 = {}, acc1 = {};
  v2f a, b0, b1;
  a.x = lds[abase]; a.y = lds[abase + 1];
  b0.x = r0[0];  b0.y = r1[0];
  b1.x = r0[16]; b1.y = r1[16];
  for (int t = 0; t < 63; ++t) {  // K=256 in chunks of 4, pipelined
    r0 += 4 * rs; r1 += 4 * rs;
    __builtin_prefetch(r0 + 28 * rs, 0, 1);  // speculative stream-ahead
    v2f an, b0n, b1n;
    an.x = lds[abase + 4 * t + 4]; an.y = lds[abase + 4 * t + 5];
    b0n.x = r0[0];  b0n.y = r1[0];
    b1n.x = r0[16]; b1n.y = r1[16];
    acc0 = wmma4(a, b0, acc0);
    acc1 = wmma4(a, b1, acc1);
    a = an; b0 = b0n; b1 = b1n;
  }
  acc0 = wmma4(a, b0, acc0);
  acc1 = wmma4(a, b1, acc1);
  if (hi == 0) {  // rows M=0..3 live in acc VGPRs 0..3, lanes 0-15
    union { v8f v; float f[8]; } u0, u1;
    u0.v = acc0; u1.v = acc1;
#pragma unroll
    for (int r = 0; r < 4; ++r) {
      Wgen[(size_t)r * N + n0 + m] = u0.f[r];
      Wgen[(size_t)r * N + n0 + 16 + m] = u1.f[r];
    }
  }
}

// Bgen[b][o] = lat_new[b] . Wb[:,o]  (tiny)
__global__ void biasgen(const float* __restrict__ Wb, const float* __restrict__ ln,
                        float* __restrict__ Bg, int fout) {
  int idx = blockIdx.x * 256 + threadIdx.x;
  if (idx >= 4 * fout) return;
  int b = idx / fout, o = idx - b * fout;
  float acc = 0.f;
  for (int k = 0; k < 256; ++k) acc += ln[b * 256 + k] * Wb[k * fout + o];
  Bg[idx] = acc;
}

// ---------------------------------------------------------------------------
// Per-sample 1x1 "dyna" conv as WMMA GEMM:
//   Out[b][o][p] = epi( sum_i W[b][o*fin+i] * X[b][i][p] + Bias[b][o] )
// 16(o) x 32(p) tile per wave (A-fragment reused by 2 WMMAs), pipelined.
// mode: 0 plain, 1 leaky-relu(0.2), 2 e*cos/e*sin split (+64ch), 3 residual.
__global__ void __launch_bounds__(256) dyna_gemm(
    const float* __restrict__ W, const float* __restrict__ Bias,
    const float* __restrict__ X, long xBStride, int fin,
    float* __restrict__ Out, int outNumCh, int outChOff, int fout, int P,
    int mode, const float* __restrict__ Skip, const float* __restrict__ leakPtr) {
  int lane = threadIdx.x & 31, wave = threadIdx.x >> 5;
  int pt = P >> 5, ot = fout >> 4;
  int tiles = 4 * ot * pt;
  int tile = blockIdx.x * WPB + wave;
  if (tile >= tiles) return;  // wave-uniform
  int b = tile / (ot * pt);
  int rem = tile - b * (ot * pt);
  int o0 = (rem / pt) << 4;
  int p0 = (rem - (rem / pt) * pt) << 5;
  int hi = lane >> 4, m = lane & 15;

  const float* Wk = W + (size_t)b * fout * fin + (size_t)(o0 + m) * fin + 2 * hi;
  const size_t rp = (size_t)P;
  const float* x0 = X + (size_t)b * xBStride + (size_t)(2 * hi) * rp + p0 + m;
  const float* x1 = x0 + rp;
  v8f acc0 = {}, acc1 = {};
  v2f a, b0, b1;
  a = *(const v2f*)Wk;
  b0.x = x0[0];  b0.y = x1[0];
  b1.x = x0[16]; b1.y = x1[16];
  for (int k0 = 0; k0 < fin - 4; k0 += 4) {
    Wk += 4; x0 += 4 * rp; x1 += 4 * rp;
    v2f an = *(const v2f*)Wk;
    v2f b0n, b1n;
    b0n.x = x0[0];  b0n.y = x1[0];
    b1n.x = x0[16]; b1n.y = x1[16];
    acc0 = wmma4(a, b0, acc0);
    acc1 = wmma4(a, b1, acc1);
    a = an; b0 = b0n; b1 = b1n;
  }
  acc0 = wmma4(a, b0, acc0);
  acc1 = wmma4(a, b1, acc1);

  float leak = 0.f;
  if (mode == 3) leak = fminf(fmaxf(leakPtr[0], 0.001f), 1000.0f);
  union { v8f v; float f[8]; } u0, u1;
  u0.v = acc0; u1.v = acc1;
  const float E1 = 2.7182818284590452f;
#pragma unroll
  for (int r = 0; r < 8; ++r) {
    int o = o0 + 8 * hi + r;
    float bias = Bias[b * fout + o];
    size_t rowbase = ((size_t)b * outNumCh + outChOff + o) * P;
#pragma unroll
    for (int q = 0; q < 2; ++q) {
      int p = p0 + 16 * q + m;
      float v = (q ? u1.f[r] : u0.f[r]) + bias;
      size_t oi = rowbase + p;
      if (mode == 0) {
        Out[oi] = v;
      } else if (mode == 1) {
        Out[oi] = v > 0.f ? v : 0.2f * v;
      } else if (mode == 2) {
        float sv, cv; sincosf(v, &sv, &cv);
        Out[oi] = E1 * cv;
        Out[oi + (size_t)64 * P] = E1 * sv;  // sin half lives +64 channels up
      } else {
        float sk = Skip[((size_t)b * fout + o) * P + p];
        Out[oi] += leak * (sk + v);
      }
    }
  }
}

// ---------------------------------------------------------------------------
__global__ void seedinit(const float* __restrict__ seed, float* __restrict__ out) {
  int t = blockIdx.x * 256 + threadIdx.x;  // 4*64*256
  out[t] = seed[t & 16383];
}

// per-(b,ch) mean over pixels  (mean commutes with the 1x1 conv for wave_c)
__global__ void chanmean(const float* __restrict__ X, float* __restrict__ Mn, int P) {
  const float* x = X + (size_t)blockIdx.x * P;
  float s = 0.f;
  for (int p = threadIdx.x; p < P; p += 256) s += x[p];
  __shared__ float red[256];
  red[threadIdx.x] = s; __syncthreads();
  for (int st = 128; st > 0; st >>= 1) {
    if (threadIdx.x < st) red[threadIdx.x] += red[threadIdx.x + st];
    __syncthreads();
  }
  if (threadIdx.x == 0) Mn[blockIdx.x] = red[0] / (float)P;
}

// wave_c[b][o] = bias + sum_i Wc2w[b][o][i] * mean_out[b][i]
__global__ void waveck(const float* __restrict__ Wg, const float* __restrict__ Bg,
                       const float* __restrict__ mn, float* __restrict__ wc) {
  int idx = threadIdx.x;
  if (idx >= 128) return;
  int b = idx >> 5, o = idx & 31;
  float acc = Bg[b * 32 + o];
  const float* w = Wg + (size_t)b * 2048 + o * 64;
  const float* m = mn + b * 64;
  for (int i = 0; i < 64; ++i) acc += w[i] * m[i];
  wc[idx] = acc;
}

// X160[b][j][p] = wave_c[b][j] * cell_wave[j][p]  (pos-enc computed on the fly)
__global__ void cellwave(const float* __restrict__ wavec, float* __restrict__ X160,
                         int s) {
  int P = s * s;
  int t = blockIdx.x * 256 + threadIdx.x;
  if (t >= 4 * 32 * P) return;
  int p = t % P;
  int rest = t / P;
  int j = rest & 31, b = rest >> 5;
  int y = p / s, x = p - y * s;
  int k = j >> 2, kind = j & 3;
  float coord = (kind < 2) ? (float)y : (float)x;
  float ang = coord * (float)(1 << k) * (6.28318530717958647f / (float)s);
  float v = (kind & 1) ? cosf(ang) : sinf(ang);
  X160[((size_t)b * 160 + j) * P + p] = wavec[b * 32 + j] * v;
}

// grads = [sobel_x(out), sobel_y(out)]  (zero-padded 3x3 cross-correlation)
__global__ void sobelk(const float* __restrict__ out, float* __restrict__ grads,
                       int s) {
  int P = s * s;
  int t = blockIdx.x * 256 + threadIdx.x;
  if (t >= 4 * 64 * P) return;
  int p = t % P, bc = t / P;
  int b = bc >> 6, c = bc & 63;
  int y = p / s, x = p - y * s;
  const float* src = out + (size_t)bc * P;
  float v[3][3];
#pragma unroll
  for (int dy = -1; dy <= 1; ++dy)
#pragma unroll
    for (int dx = -1; dx <= 1; ++dx) {
      int yy = y + dy, xx = x + dx;
      v[dy + 1][dx + 1] =
          (yy < 0 || yy >= s || xx < 0 || xx >= s) ? 0.f : src[yy * s + xx];
    }
  float gx = (v[0][2] - v[0][0] + 2.f * (v[1][2] - v[1][0]) + v[2][2] - v[2][0]) * 0.125f;
  float gy = (v[2][0] - v[0][0] + 2.f * (v[2][1] - v[0][1]) + v[2][2] - v[0][2]) * 0.125f;
  grads[((size_t)b * 128 + c) * P + p] = gx;
  grads[((size_t)b * 128 + 64 + c) * P + p] = gy;
}

// instance norm, in place, per (b, ch)
__global__ void inormk(float* __restrict__ X, int P) {
  float* x = X + (size_t)blockIdx.x * P;
  float s = 0.f, s2 = 0.f;
  for (int p = threadIdx.x; p < P; p += 256) { float v = x[p]; s += v; s2 += v * v; }
  __shared__ float r1[256], r2[256];
  r1[threadIdx.x] = s; r2[threadIdx.x] = s2; __syncthreads();
  for (int st = 128; st > 0; st >>= 1) {
    if (threadIdx.x < st) {
      r1[threadIdx.x] += r1[threadIdx.x + st];
      r2[threadIdx.x] += r2[threadIdx.x + st];
    }
    __syncthreads();
  }
  float mean = r1[0] / (float)P;
  float var = r2[0] / (float)P - mean * mean;
  float rs = rsqrtf(var + 1e-5f);
  for (int p = threadIdx.x; p < P; p += 256) x[p] = (x[p] - mean) * rs;
}

// bilinear 2x upsample (half-pixel centers, edge clamp)
__global__ void resize2x(const float* __restrict__ src, float* __restrict__ dst,
                         int s) {
  int S = 2 * s, P2 = S * S;
  int t = blockIdx.x * 256 + threadIdx.x;
  if (t >= 4 * 64 * P2) return;
  int p = t % P2, bc = t / P2;
  int y = p / S, x = p - y * S;
  float sy = (float)y * 0.5f - 0.25f;
  float sx = (float)x * 0.5f - 0.25f;
  int y0 = (int)floorf(sy), x0 = (int)floorf(sx);
  float fy = sy - (float)y0, fx = sx - (float)x0;
  int y0c = min(max(y0, 0), s - 1), y1c = min(max(y0 + 1, 0), s - 1);
  int x0c = min(max(x0, 0), s - 1), x1c = min(max(x0 + 1, 0), s - 1);
  const float* sp = src + (size_t)bc * s * s;
  float v00 = sp[y0c * s + x0c], v01 = sp[y0c * s + x1c];
  float v10 = sp[y1c * s + x0c], v11 = sp[y1c * s + x1c];
  dst[(size_t)bc * P2 + p] =
      (1.f - fy) * ((1.f - fx) * v00 + fx * v01) +
      fy * ((1.f - fx) * v10 + fx * v11);
}

// gaussian 3x3 depthwise (sigma=1 kernel, normalized, zero pad)
__global__ void gauss3(const float* __restrict__ src, float* __restrict__ dst,
                       int s) {
  int P = s * s;
  int t = blockIdx.x * 256 + threadIdx.x;
  if (t >= 4 * 64 * P) return;
  int p = t % P, bc = t / P;
  int y = p / s, x = p - y * s;
  const float gq = 0.60653065971263342f;           // exp(-0.5)
  const float inorm = 1.f / ((1.f + 2.f * gq) * (1.f + 2.f * gq));
  const float* sp = src + (size_t)bc * P;
  float acc = 0.f;
#pragma unroll
  for (int dy = -1; dy <= 1; ++dy)
#pragma unroll
    for (int dx = -1; dx <= 1; ++dx) {
      int yy = y + dy, xx = x + dx;
      float w = ((dy == 0) ? 1.f : gq) * ((dx == 0) ? 1.f : gq);
      acc += w * ((yy < 0 || yy >= s || xx < 0 || xx >= s) ? 0.f : sp[yy * s + xx]);
    }
  dst[(size_t)bc * P + p] = acc * inorm;
}

// img = clip(img_W @ out + img_b, -1, 1)
__global__ void imgk(const float* __restrict__ out, const float* __restrict__ W,
                     const float* __restrict__ Bi, float* __restrict__ dst) {
  int t = blockIdx.x * 256 + threadIdx.x;  // 4*4096
  int p = t & 4095, b = t >> 12;
  float a0 = Bi[0], a1 = Bi[1], a2 = Bi[2];
  const float* op = out + (size_t)b * 64 * 4096 + p;
  for (int i = 0; i < 64; ++i) {
    float v = op[(size_t)i * 4096];
    a0 += W[i] * v; a1 += W[64 + i] * v; a2 += W[128 + i] * v;
  }
  a0 = fminf(fmaxf(a0, -1.f), 1.f);
  a1 = fminf(fmaxf(a1, -1.f), 1.f);
  a2 = fminf(fmaxf(a2, -1.f), 1.f);
  dst[(size_t)b * 12288 + p] = a0;
  dst[(size_t)b * 12288 + 4096 + p] = a1;
  dst[(size_t)b * 12288 + 8192 + p] = a2;
}

// ---------------------------------------------------------------------------
extern "C" void kernel_launch(void* const* d_in, const int* in_sizes, int n_in,
                              void* d_out, int out_size, void* d_ws, size_t ws_size,
                              hipStream_t stream) {
  (void)in_sizes; (void)n_in; (void)out_size; (void)ws_size;
  const float* lat = (const float*)d_in[0];
  const float* leakP = (const float*)d_in[1];
  const float* seed = (const float*)d_in[2];
  const float* wbias[3] = {(const float*)d_in[3], (const float*)d_in[4],
                           (const float*)d_in[5]};
  const float* fracW = (const float*)d_in[6];
  const float* fracB = (const float*)d_in[7];
  const float* imgW = (const float*)d_in[24];
  const float* imgB = (const float*)d_in[25];

  // modules: bt, c2w, w2c, g2c, cc_skip, cc_c1, cc_c2, cc_c3
  const int foutA[8] = {64, 32, 128, 128, 64, 128, 128, 64};
  const int finA[8]  = {1024, 64, 160, 128, 256, 256, 128, 128};
  const float* WwP[8] = {(const float*)d_in[8],  (const float*)d_in[10],
                         (const float*)d_in[12], (const float*)d_in[14],
                         (const float*)d_in[16], (const float*)d_in[18],
                         (const float*)d_in[20], (const float*)d_in[22]};
  const float* WbP[8] = {(const float*)d_in[9],  (const float*)d_in[11],
                         (const float*)d_in[13], (const float*)d_in[15],
                         (const float*)d_in[17], (const float*)d_in[19],
                         (const float*)d_in[21], (const float*)d_in[23]};

  float* ws = (float*)d_ws;
  size_t off = 0;
  auto take = [&](size_t n) {
    float* p = ws + off;
    off += (n + 255) & ~(size_t)255;
    return p;
  };
  float* lnew = take(1024);
  float* wgv[8]; float* bgv[8];
  for (int i = 0; i < 8; ++i) wgv[i] = take((size_t)4 * foutA[i] * finA[i]);
  for (int i = 0; i < 8; ++i) bgv[i] = take((size_t)4 * foutA[i]);
  float* outA = take((size_t)4 * 64 * 4096);
  float* outB = take((size_t)4 * 64 * 4096);
  float* X160 = take((size_t)4 * 160 * 4096);  // [wave_c | e*cos | e*sin]
  float* grd  = take((size_t)4 * 128 * 4096);
  float* hbuf = take((size_t)4 * 256 * 4096);  // [grads_enc | wave_enc]
  float* a1   = take((size_t)4 * 128 * 4096);  // also resize tmp
  float* a2   = take((size_t)4 * 128 * 4096);
  float* skb  = take((size_t)4 * 64 * 4096);
  float* mnb  = take(256);
  float* wcb  = take(128);

  seedinit<<<256, 256, 0, stream>>>(seed, outA);
  float* outCur = outA;
  float* outNxt = outB;

  for (int c = 0; c < 6; ++c) {
    int l = c >> 1;
    int s = 16 << l;
    int P = s * s;
    if ((c & 1) == 0) {
      latnewk<<<1, 256, 0, stream>>>(lat, fracW + (size_t)l * 65536,
                                     fracB + (size_t)l * 256, lnew);
      for (int i = 0; i < 8; ++i) {
        int N = foutA[i] * finA[i];
        weightgen<<<N / 256, 256, 0, stream>>>(WwP[i] + (size_t)l * 256 * N, lnew,
                                               wgv[i], N);
        biasgen<<<(4 * foutA[i] + 255) / 256, 256, 0, stream>>>(
            WbP[i] + (size_t)l * 256 * foutA[i], lnew, bgv[i], foutA[i]);
      }
      // surf GEMM + fused exp(1+i*surf): writes X160 channels 32..159
      int tiles = 4 * (64 / 16) * (P / 32);
      dyna_gemm<<<tiles / WPB, 256, 0, stream>>>(wgv[0], bgv[0], wbias[l], (long)0,
                                                 1024, X160, 160, 32, 64, P, 2,
                                                 nullptr, nullptr);
    }
    chanmean<<<256, 256, 0, stream>>>(outCur, mnb, P);
    waveck<<<1, 128, 0, stream>>>(wgv[1], bgv[1], mnb, wcb);
    cellwave<<<(4 * 32 * P) / 256, 256, 0, stream>>>(wcb, X160, s);
    {  // wave_enc -> h channels 128..255
      int tiles = 4 * (128 / 16) * (P / 32);
      dyna_gemm<<<tiles / WPB, 256, 0, stream>>>(wgv[2], bgv[2], X160,
                                                 (long)160 * P, 160, hbuf, 256,
                                                 128, 128, P, 0, nullptr, nullptr);
    }
    sobelk<<<(4 * 64 * P) / 256, 256, 0, stream>>>(outCur, grd, s);
    {  // grads_enc -> h channels 0..127
      int tiles = 4 * (128 / 16) * (P / 32);
      dyna_gemm<<<tiles / WPB, 256, 0, stream>>>(wgv[3], bgv[3], grd,
                                                 (long)128 * P, 128, hbuf, 256, 0,
                                                 128, P, 0, nullptr, nullptr);
    }
    inormk<<<1024, 256, 0, stream>>>(hbuf, P);  // inorm both halves of h
    {  // skip
      int tiles = 4 * (64 / 16) * (P / 32);
      dyna_gemm<<<tiles / WPB, 256, 0, stream>>>(wgv[4], bgv[4], hbuf,
                                                 (long)256 * P, 256, skb, 64, 0,
                                                 64, P, 0, nullptr, nullptr);
    }
    {  // c1 + lrelu
      int tiles = 4 * (128 / 16) * (P / 32);
      dyna_gemm<<<tiles / WPB, 256, 0, stream>>>(wgv[5], bgv[5], hbuf,
                                                 (long)256 * P, 256, a1, 128, 0,
                                                 128, P, 1, nullptr, nullptr);
    }
    {  // c2 + lrelu
      int tiles = 4 * (128 / 16) * (P / 32);
      dyna_gemm<<<tiles / WPB, 256, 0, stream>>>(wgv[6], bgv[6], a1,
                                                 (long)128 * P, 128, a2, 128, 0,
                                                 128, P, 1, nullptr, nullptr);
    }
    {  // c3 + residual: out += clamp(leak) * (skip + c3)
      int tiles = 4 * (64 / 16) * (P / 32);
      dyna_gemm<<<tiles / WPB, 256, 0, stream>>>(wgv[7], bgv[7], a2,
                                                 (long)128 * P, 128, outCur, 64, 0,
                                                 64, P, 3, skb, leakP);
    }
    if (c == 1 || c == 3) {
      int S = 2 * s, P2 = S * S;
      resize2x<<<(4 * 64 * P2) / 256, 256, 0, stream>>>(outCur, a1, s);
      gauss3<<<(4 * 64 * P2) / 256, 256, 0, stream>>>(a1, outNxt, S);
      float* t = outCur; outCur = outNxt; outNxt = t;
    }
  }
  imgk<<<64, 256, 0, stream>>>(outCur, imgW, imgB, (float*)d_out);
}